// GIGP_18485539241991
// MI455X (gfx1250) — compile-verified
//
#include <hip/hip_runtime.h>
#include <hip/hip_bf16.h>

// Problem constants (from reference): S=16, M=256, NF=4096, OUT=256
#define S_    16
#define M_    256
#define NF_   4096
#define OUT_  256
#define K_    256                          // every GEMM in this pipeline has K = M_ = 256
#define MM_   (M_ * M_)                    // 65536
#define MNF_  (M_ * NF_)                   // 1048576
#define OROW_ (M_ + NF_)                   // 4352 rows per batch in d_out
#define OSTR_ ((long long)OROW_ * OUT_)    // 1114112 floats per batch in d_out

typedef float v2f __attribute__((ext_vector_type(2)));
typedef float v8f __attribute__((ext_vector_type(8)));
typedef unsigned int u32x4 __attribute__((ext_vector_type(4)));
typedef int i32x4 __attribute__((ext_vector_type(4)));
typedef int i32x8 __attribute__((ext_vector_type(8)));

// ---------------------------------------------------------------------------
// TDM: DMA a dense row-major 256x256 fp32 tile from global memory into LDS
// (offset lds_byte_off in this workgroup's dynamic LDS), per CDNA5 ISA
// D# layout. Issued by one wave; caller must pass a wave-uniform condition.
// Completion via TENSORcnt, then the caller's __syncthreads publishes it.
// ---------------------------------------------------------------------------
__device__ __forceinline__ void tdm_load_256x256(const float* gptr, unsigned lds_byte_off) {
#if defined(__HIP_DEVICE_COMPILE__)
  const unsigned long long ga = (unsigned long long)(size_t)gptr;
  u32x4 g0;
  g0.x = 1u;                                                 // count=1 (valid user D#)
  g0.y = lds_byte_off;                                       // lds_addr (bytes)
  g0.z = (unsigned)(ga & 0xffffffffu);                       // global_addr[31:0]
  g0.w = (unsigned)((ga >> 32) & 0x1ffffffu) | (2u << 30);   // global_addr[56:32] | type=2
  i32x8 g1;
  g1[0] = (int)(2u << 16);        // workgroup_mask=0 (no cluster), data_size=4B
  g1[1] = (int)(256u << 16);      // tensor_dim0 = 256 (bits 79:48, low half)
  g1[2] = (int)(256u << 16);      // tensor_dim0 hi=0 | tensor_dim1 = 256 (low half)
  g1[3] = (int)(256u << 16);      // tensor_dim1 hi=0 | tile_dim0 = 256
  g1[4] = 256;                    // tile_dim1 = 256, tile_dim2 = 0
  g1[5] = 256;                    // tensor_dim0_stride = 256 (low 32)
  g1[6] = 0;                      // stride0 hi16 = 0 | tensor_dim1_stride low16 = 0
  g1[7] = 1;                      // tensor_dim1_stride = 65536 (bits 47:16 -> 1)
  const i32x4 z4 = {0, 0, 0, 0};  // 2D tensor: groups 2/3 unused
#if __clang_major__ >= 23
  const i32x8 z8 = {0, 0, 0, 0, 0, 0, 0, 0};
  __builtin_amdgcn_tensor_load_to_lds(g0, g1, z4, z4, z8, 0);
#else
  __builtin_amdgcn_tensor_load_to_lds(g0, g1, z4, z4, 0);
#endif
  __builtin_amdgcn_s_wait_tensorcnt(0);
#else
  (void)gptr; (void)lds_byte_off;
#endif
}

// ---------------------------------------------------------------------------
// Scalar prep: scal[0] = 1e-8 * max(Kuu); scal[1] = exp(L_scale)/mean(diag(L_loc))
// ---------------------------------------------------------------------------
__global__ void scalars_kernel(const float* __restrict__ Kuu, int nK,
                               const float* __restrict__ L_loc,
                               const float* __restrict__ L_scale,
                               float* __restrict__ scal) {
  __shared__ float red[256];
  const int t = threadIdx.x;
  float mx = -3.4e38f;
  for (int i = t; i < nK; i += 256) mx = fmaxf(mx, Kuu[i]);
  red[t] = mx;
  __syncthreads();
  for (int w = 128; w > 0; w >>= 1) {
    if (t < w) red[t] = fmaxf(red[t], red[t + w]);
    __syncthreads();
  }
  const float maxk = red[0];
  __syncthreads();
  red[t] = L_loc[t * M_ + t];  // blockDim == M_ == 256
  __syncthreads();
  for (int w = 128; w > 0; w >>= 1) {
    if (t < w) red[t] += red[t + w];
    __syncthreads();
  }
  if (t == 0) {
    scal[0] = 1e-8f * maxk;
    scal[1] = __expf(L_scale[0]) / (red[0] / (float)M_);
  }
}

__global__ void scale_L_kernel(const float* __restrict__ L_loc,
                               const float* __restrict__ scal,
                               float* __restrict__ L) {
  const int i = blockIdx.x * 256 + threadIdx.x;  // grid covers MM_ exactly
  L[i] = L_loc[i] * scal[1];
}

// ---------------------------------------------------------------------------
// Batched 256x256 Cholesky, matrix resident in LDS (256 KB of the 320 KB WGP LDS).
// Matrix staged by the Tensor Data Mover (wave 0), factorized by 256 threads
// (thread t owns row t). One workgroup per batch matrix.
// ---------------------------------------------------------------------------
__global__ void cholesky256(const float* __restrict__ Ain, long long sA,
                            float* __restrict__ Lout, long long sL,
                            const float* __restrict__ jit) {
  extern __shared__ float sh[];  // MM_ floats at dynamic-LDS offset 0
  const int s = blockIdx.x, t = threadIdx.x;
  const float* __restrict__ Ab = Ain + (long long)s * sA;
  if (t < 32) tdm_load_256x256(Ab, 0u);   // wave-uniform guard; TDM -> LDS
  __syncthreads();
  if (jit) sh[t * M_ + t] += jit[0];
  __syncthreads();
  for (int k = 0; k < M_; ++k) {
    if (t == k) sh[k * M_ + k] = sqrtf(sh[k * M_ + k]);
    __syncthreads();
    if (t > k) sh[t * M_ + k] /= sh[k * M_ + k];
    __syncthreads();
    if (t > k) {
      const float lik = sh[t * M_ + k];
      for (int j = k + 1; j <= t; ++j) sh[t * M_ + j] -= lik * sh[j * M_ + k];
    }
  }
  __syncthreads();
  float* __restrict__ Lb = Lout + (long long)s * sL;
  for (int i = t; i < MM_; i += 256) Lb[i] = sh[i];  // upper part unused downstream
}

// ---------------------------------------------------------------------------
// Triangular solve: op(L) X = B, L is 256x256 lower Cholesky factor staged by
// TDM into LDS; RHS processed in 32-column panels (256KB + 32KB < 320KB LDS).
// backward=0: L Y = B (forward);  backward=1: L^T X = B (uses L[k][i], i<k).
// transB: load RHS element (r,c) from Bin^T (i.e. Bin[c*ldb + r]).
// ---------------------------------------------------------------------------
__global__ void trsm256(const float* __restrict__ Lc, long long sL,
                        const float* __restrict__ Bin, long long sB, int ldb, int transB,
                        float* __restrict__ X, long long sX, int ldx,
                        int ncols, int backward) {
  extern __shared__ float sm[];
  float* shL = sm;        // MM_ floats (LDS byte offset 0)
  float* shB = sm + MM_;  // M_ * 32 floats
  const int s = blockIdx.y;
  const int c0 = blockIdx.x * 32;
  const int t = threadIdx.x;  // 256 threads
  const float* __restrict__ Lb = Lc + (long long)s * sL;
  if (t < 32) tdm_load_256x256(Lb, 0u);   // factor via Tensor Data Mover
  const float* __restrict__ Bb = Bin + (long long)s * sB;
  for (int i = t; i < M_ * 32; i += 256) {
    const int r = i >> 5, c = i & 31;
    const int gc = c0 + c;
    float v = 0.f;
    if (gc < ncols)
      v = transB ? Bb[(long long)gc * ldb + r] : Bb[(long long)r * ldb + gc];
    shB[i] = v;
  }
  __syncthreads();
  if (!backward) {
    for (int k = 0; k < M_; ++k) {
      if (t < 32) shB[(k << 5) + t] /= shL[k * M_ + k];
      __syncthreads();
      const int rows = M_ - 1 - k;
      for (int i = t; i < rows * 32; i += 256) {
        const int rr = k + 1 + (i >> 5), c = i & 31;
        shB[(rr << 5) + c] -= shL[rr * M_ + k] * shB[(k << 5) + c];
      }
      __syncthreads();
    }
  } else {
    for (int k = M_ - 1; k >= 0; --k) {
      if (t < 32) shB[(k << 5) + t] /= shL[k * M_ + k];
      __syncthreads();
      for (int i = t; i < k * 32; i += 256) {
        const int rr = i >> 5, c = i & 31;
        shB[(rr << 5) + c] -= shL[k * M_ + rr] * shB[(k << 5) + c];
      }
      __syncthreads();
    }
  }
  float* __restrict__ Xb = X + (long long)s * sX;
  for (int i = t; i < M_ * 32; i += 256) {
    const int r = i >> 5, c = i & 31;
    const int gc = c0 + c;
    if (gc < ncols) Xb[(long long)r * ldx + gc] = shB[i];
  }
}

// ---------------------------------------------------------------------------
// fp32 WMMA GEMM (K fixed = 256):
//   C = alpha * op(A) @ op(B) + beta * C0 + gamma * I
// TA/TB are compile-time -> straight-line inner loop, no branches.
//
// Block = 128 threads (4 waves). Block computes a 64x16 tile of C:
//   - B panel (256x16) staged once in LDS, shared by all 4 waves (4x reuse)
//   - per-wave A panel (16x256) staged coalesced
// Panels stored pair-interleaved [(k/2)*32 + m*2 + (k&1)] so each lane's WMMA
// fragment is one aligned ds_load_b64; lanes 0-15 hit banks 0-31 and lanes
// 16-31 hit banks 32-63 -> conflict-free across all 64 LDS banks.
// Inner loop: 2x ds_load_b64 + 1x v_wmma_f32_16x16x4_f32, unrolled x8.
// ---------------------------------------------------------------------------
#define GEMM_LDS_BYTES ((K_ * 16 + 4 * 16 * K_) * 4)   // 16KB B + 64KB A = 80KB

template <int TA, int TB>
__global__ void gemm_wmma_t(const float* __restrict__ A, const float* __restrict__ B,
                            const float* __restrict__ C0, float* __restrict__ C,
                            int lda, int ldb, int ldc0, int ldc,
                            long long sA, long long sB, long long sC0, long long sC,
                            float alpha, float beta, float gamma) {
  extern __shared__ float sm[];
  float* shB = sm;             // K_ * 16
  float* shA = sm + K_ * 16;   // 4 * 16 * K_
  const int tid  = threadIdx.x;
  const int s    = blockIdx.z;
  const int wave = tid >> 5;
  const int lane = tid & 31;
  const int half = lane >> 4;
  const int lm   = lane & 15;
  const int mBase = blockIdx.y * 64;   // grids sized exactly: Mdim % 64 == 0
  const int n0    = blockIdx.x * 16;   // Ndim % 16 == 0

  const float* __restrict__ Ab = A + (long long)s * sA;
  const float* __restrict__ Bb = B + (long long)s * sB;

  // ---- stage B panel (K_ x 16 at column n0), coalesced per layout ----
  if (TB == 0) {
    for (int i = tid; i < K_ * 16; i += 128) {
      const int k = i >> 4, n = i & 15;
      shB[(k >> 1) * 32 + n * 2 + (k & 1)] = Bb[(long long)k * ldb + (n0 + n)];
    }
  } else {
    for (int i = tid; i < K_ * 16; i += 128) {
      const int n = i >> 8, k = i & (K_ - 1);
      shB[(k >> 1) * 32 + n * 2 + (k & 1)] = Bb[(long long)(n0 + n) * ldb + k];
    }
  }
  // ---- stage A panels (64 rows at mBase x K_), coalesced per layout ----
  if (TA == 0) {
    for (int i = tid; i < 64 * K_; i += 128) {
      const int m = i >> 8, k = i & (K_ - 1);
      shA[(m >> 4) * (16 * K_) + (k >> 1) * 32 + (m & 15) * 2 + (k & 1)] =
          Ab[(long long)(mBase + m) * lda + k];
    }
  } else {
    for (int i = tid; i < 64 * K_; i += 128) {
      const int k = i >> 6, m = i & 63;
      shA[(m >> 4) * (16 * K_) + (k >> 1) * 32 + (m & 15) * 2 + (k & 1)] =
          Ab[(long long)k * lda + (mBase + m)];
    }
  }

  const long long cb  = (long long)s * sC;
  const long long c0b = (long long)s * sC0;
  const int m0 = mBase + wave * 16;
  if (C0) __builtin_prefetch(C0 + c0b + (long long)m0 * ldc0 + n0, 0, 0);

  __syncthreads();

  const float* __restrict__ sAw = shA + wave * (16 * K_);
  const int fo = lm * 2;  // fragment lane offset within a 32-float pair-row
  v8f acc = {};
#pragma unroll 8
  for (int k0 = 0; k0 < K_; k0 += 4) {
    const int kr = (k0 >> 1) + half;  // (ka/2) with ka = k0 + 2*half (even)
    const v2f a = *(const v2f*)(sAw + kr * 32 + fo);
    const v2f b = *(const v2f*)(shB + kr * 32 + fo);
    acc = __builtin_amdgcn_wmma_f32_16x16x4_f32(false, a, false, b,
                                                (short)0, acc, false, false);
  }

#pragma unroll
  for (int r = 0; r < 8; ++r) {
    const int mi = m0 + r + 8 * half;
    const int ni = n0 + lm;
    float v = alpha * acc[r];
    if (C0) v += beta * C0[c0b + (long long)mi * ldc0 + ni];
    if (gamma != 0.f && mi == ni) v += gamma;
    C[cb + (long long)mi * ldc + ni] = v;
  }
}

static inline void launch_gemm(int TA, int TB, dim3 grid, hipStream_t st,
                               const float* A, const float* B, const float* C0, float* C,
                               int lda, int ldb, int ldc0, int ldc,
                               long long sA, long long sB, long long sC0, long long sC,
                               float alpha, float beta, float gamma) {
  if (TA == 0 && TB == 0)
    gemm_wmma_t<0, 0><<<grid, 128, GEMM_LDS_BYTES, st>>>(A, B, C0, C, lda, ldb, ldc0, ldc,
                                                         sA, sB, sC0, sC, alpha, beta, gamma);
  else if (TA == 1 && TB == 0)
    gemm_wmma_t<1, 0><<<grid, 128, GEMM_LDS_BYTES, st>>>(A, B, C0, C, lda, ldb, ldc0, ldc,
                                                         sA, sB, sC0, sC, alpha, beta, gamma);
  else if (TA == 0 && TB == 1)
    gemm_wmma_t<0, 1><<<grid, 128, GEMM_LDS_BYTES, st>>>(A, B, C0, C, lda, ldb, ldc0, ldc,
                                                         sA, sB, sC0, sC, alpha, beta, gamma);
  else
    gemm_wmma_t<1, 1><<<grid, 128, GEMM_LDS_BYTES, st>>>(A, B, C0, C, lda, ldb, ldc0, ldc,
                                                         sA, sB, sC0, sC, alpha, beta, gamma);
}

// ---------------------------------------------------------------------------
// VMAT[s] += t2 (broadcast the shared L @ (L^T @ u_param) term over batches)
// ---------------------------------------------------------------------------
__global__ void add_bcast(float* __restrict__ V, const float* __restrict__ t2) {
  const int idx = blockIdx.x * 256 + threadIdx.x;  // grid covers S_*MM_ exactly
  V[idx] += t2[idx & (MM_ - 1)];
}

// ---------------------------------------------------------------------------
// Vf[s][j] = Kff_diag[s][j] - sum_i AW[s][i][j] * Kuf[s][i][j]   (coalesced in j)
// ---------------------------------------------------------------------------
__global__ void vf_reduce(const float* __restrict__ AW, const float* __restrict__ Kuf,
                          const float* __restrict__ Kffd, float* __restrict__ Vf) {
  const int s = blockIdx.y;
  const int j = blockIdx.x * 256 + threadIdx.x;
  const long long base = (long long)s * MNF_;
  float acc = 0.f;
  for (int i = 0; i < M_; ++i)
    acc = fmaf(AW[base + (long long)i * NF_ + j], Kuf[base + (long long)i * NF_ + j], acc);
  Vf[s * NF_ + j] = Kffd[s * NF_ + j] - acc;
}

// ---------------------------------------------------------------------------
// f region of d_out:  out[s][M+j][o] += sqrt(Vf[s][j]) * noise_f[s][j][o]
// ---------------------------------------------------------------------------
__global__ void f_finalize(float* __restrict__ out, const float* __restrict__ Vf,
                           const float* __restrict__ noise_f, long long total) {
  const long long idx = (long long)blockIdx.x * 256 + threadIdx.x;
  if (idx >= total) return;
  const int o = (int)(idx % OUT_);
  const long long jj = idx / OUT_;
  const int j = (int)(jj % NF_);
  const int s = (int)(jj / NF_);
  const float sv = sqrtf(fmaxf(Vf[s * NF_ + j], 0.f));
  out[(long long)s * OSTR_ + (long long)(M_ + j) * OUT_ + o] += sv * noise_f[idx];
}

// ---------------------------------------------------------------------------
// Orchestration
// ---------------------------------------------------------------------------
extern "C" void kernel_launch(void* const* d_in, const int* in_sizes, int n_in,
                              void* d_out, int out_size, void* d_ws, size_t ws_size,
                              hipStream_t stream) {
  const float* Kuu   = (const float*)d_in[0];  // (S,M,M)
  const float* Kuf   = (const float*)d_in[1];  // (S,M,NF)
  const float* Kffd  = (const float*)d_in[2];  // (S,NF)
  const float* L_loc = (const float*)d_in[3];  // (M,M)
  const float* L_sc  = (const float*)d_in[4];  // scalar
  const float* u_par = (const float*)d_in[5];  // (OUT,M,1)
  const float* n_inv = (const float*)d_in[6];  // (S,OUT,M,1)
  const float* n_L   = (const float*)d_in[7];  // (S,OUT,M,1)
  const float* n_f   = (const float*)d_in[8];  // (S,NF,OUT)
  float* out = (float*)d_out;                  // (S, M+NF, OUT)

  // Workspace layout (floats), ~93 MB total
  float* ws   = (float*)d_ws;
  float* SCAL = ws;                             // 64      : [0]=jitter, [1]=c
  float* Lw   = SCAL + 64;                      // MM_     : L = c * L_loc
  float* CHK  = Lw + MM_;                       // S_*MM_  : chol(Kuu_j)
  float* KUL  = CHK + (size_t)S_ * MM_;         // S_*MM_  : Kuu @ L
  float* T1   = KUL + (size_t)S_ * MM_;         // S_*MM_  : L^T KuuL + I -> chol_l
  float* Wws  = T1 + (size_t)S_ * MM_;          // S_*MM_  : cho_solve(chol_l, KuuL^T)
  float* SIG  = Wws + (size_t)S_ * MM_;         // S_*MM_  : Sigma
  float* VMAT = SIG + (size_t)S_ * MM_;         // S_*MM_  : prec_noise + L(L^T u)
  float* TU   = VMAT + (size_t)S_ * MM_;        // MM_     : L^T @ u_param^T
  float* T2   = TU + MM_;                       // MM_     : L @ TU
  float* VF   = T2 + MM_;                       // S_*NF_  : Vf
  float* AW   = VF + (size_t)S_ * NF_;          // S_*MNF_ : Kuu^{-1} Kuf (64 MB)

  const size_t chol_lds = (size_t)MM_ * 4;                   // 256 KB
  const size_t trsm_lds = (size_t)(MM_ + M_ * 32) * 4;       // 288 KB (< 320 KB WGP LDS)

  // --- scalars + scaled L ---
  scalars_kernel<<<1, 256, 0, stream>>>(Kuu, S_ * MM_, L_loc, L_sc, SCAL);
  scale_L_kernel<<<MM_ / 256, 256, 0, stream>>>(L_loc, SCAL, Lw);

  // --- chol(Kuu + jit*I) per batch ---
  cholesky256<<<S_, 256, chol_lds, stream>>>(Kuu, MM_, CHK, MM_, SCAL);

  // --- KuuL = Kuu @ L ---
  launch_gemm(0, 0, dim3(16, 4, S_), stream, Kuu, Lw, nullptr, KUL,
              M_, M_, 0, M_, MM_, 0, 0, MM_, 1.f, 0.f, 0.f);

  // --- T1 = L^T @ KuuL + I ; chol_l (in place) ---
  launch_gemm(1, 0, dim3(16, 4, S_), stream, Lw, KUL, nullptr, T1,
              M_, M_, 0, M_, 0, MM_, 0, MM_, 1.f, 0.f, 1.f);
  cholesky256<<<S_, 256, chol_lds, stream>>>(T1, MM_, T1, MM_, nullptr);

  // --- W = cho_solve(chol_l, KuuL^T): forward (trans-load RHS) then backward in place ---
  trsm256<<<dim3(M_ / 32, S_), 256, trsm_lds, stream>>>(T1, MM_, KUL, MM_, M_, 1,
      Wws, MM_, M_, M_, 0);
  trsm256<<<dim3(M_ / 32, S_), 256, trsm_lds, stream>>>(T1, MM_, Wws, MM_, M_, 0,
      Wws, MM_, M_, M_, 1);

  // --- Sigma = Kuu - KuuL @ W ---
  launch_gemm(0, 0, dim3(16, 4, S_), stream, KUL, Wws, Kuu, SIG,
              M_, M_, M_, M_, MM_, MM_, MM_, MM_, -1.f, 1.f, 0.f);

  // --- t_u = L^T @ u_param^T ; t2 = L @ t_u (shared across batches) ---
  launch_gemm(1, 1, dim3(16, 4, 1), stream, Lw, u_par, nullptr, TU,
              M_, M_, 0, OUT_, 0, 0, 0, 0, 1.f, 0.f, 0.f);
  launch_gemm(0, 0, dim3(16, 4, 1), stream, Lw, TU, nullptr, T2,
              M_, OUT_, 0, OUT_, 0, 0, 0, 0, 1.f, 0.f, 0.f);

  // --- VMAT = chol_Kuu^T \ z  (single backward solve; z = noise_inv^T per batch) ---
  trsm256<<<dim3(OUT_ / 32, S_), 256, trsm_lds, stream>>>(CHK, MM_, n_inv, MM_, M_, 1,
      VMAT, MM_, OUT_, OUT_, 1);

  // --- VMAT += L @ noise_L^T  (C0 = VMAT, beta = 1) ; VMAT += t2 ---
  launch_gemm(0, 1, dim3(16, 4, S_), stream, Lw, n_L, VMAT, VMAT,
              M_, M_, OUT_, OUT_, 0, MM_, MM_, MM_, 1.f, 1.f, 0.f);
  add_bcast<<<(S_ * MM_) / 256, 256, 0, stream>>>(VMAT, T2);

  // --- u = Sigma @ VMAT, written straight into d_out rows [0, M) of each batch ---
  launch_gemm(0, 0, dim3(16, 4, S_), stream, SIG, VMAT, nullptr, out,
              M_, OUT_, 0, OUT_, MM_, MM_, 0, OSTR_, 1.f, 0.f, 0.f);

  // --- AW = Kuu^{-1} Kuf: forward then backward (in place), 4096 RHS in 32-col panels ---
  trsm256<<<dim3(NF_ / 32, S_), 256, trsm_lds, stream>>>(CHK, MM_, Kuf, MNF_, NF_, 0,
      AW, MNF_, NF_, NF_, 0);
  trsm256<<<dim3(NF_ / 32, S_), 256, trsm_lds, stream>>>(CHK, MM_, AW, MNF_, NF_, 0,
      AW, MNF_, NF_, NF_, 1);

  // --- Ef = AW^T @ u, written straight into d_out rows [M, M+NF) ---
  launch_gemm(1, 0, dim3(16, 64, S_), stream, AW, out, nullptr, out + (size_t)M_ * OUT_,
              NF_, OUT_, 0, OUT_, MNF_, OSTR_, 0, OSTR_, 1.f, 0.f, 0.f);

  // --- Vf and final f = Ef + sqrt(Vf) * noise_f ---
  vf_reduce<<<dim3(NF_ / 256, S_), 256, 0, stream>>>(AW, Kuf, Kffd, VF);
  const long long tot = (long long)S_ * NF_ * OUT_;
  f_finalize<<<(unsigned)((tot + 255) / 256), 256, 0, stream>>>(out, VF, n_f, tot);
}